// HeteroGNN_28200755266017
// MI455X (gfx1250) — compile-verified
//
#include <hip/hip_runtime.h>

// ---------------------------------------------------------------------------
// HeteroGNN (2-layer SAGE, aggr='add') for MI455X / gfx1250.
//
// Phase 1 (per relation): scatter-add source rows into agg  (L2-resident,
//   global_atomic_add_f32, one wave32 per edge, float2 per lane).
// Phase 2 (per relation): acc[dst] += agg @ Wl.T + bl + x[dst] @ Wr.T using
//   V_WMMA_F32_16X16X4_F32 (full fp32 precision; GEMM phase is ~8.5 GFLOP
//   total, negligible vs the ~1 GB/layer scatter traffic).
// Then leaky_relu per node type; final ticker @ lin_W.T + lin_b via the same
// WMMA kernel with Nout=32.
//
// Param leaf order (jax pytree, dict keys sorted):
//   d_in[0..7]: x_ticker, x_news, x_inst, ei_mentions, ei_mentioned_by,
//               ei_holds, ei_held_by, ei_related
//   d_in[8..37]: layers[0..1] x rels{held_by,holds,mentioned_by,mentions,
//               related} x (Wl, Wr, bl)
//   d_in[38]: lin_W (32x64), d_in[39]: lin_b (32)
// ---------------------------------------------------------------------------

typedef __attribute__((ext_vector_type(2))) float v2f;
typedef __attribute__((ext_vector_type(8))) float v8f;

#define N_TICKER 50000
#define N_NEWS   100000
#define N_INST   10000

// ---- zero fill (float4 granularity) ---------------------------------------
__global__ __launch_bounds__(256) void k_zero(float* __restrict__ p, int n4) {
  int i = blockIdx.x * 256 + threadIdx.x;
  if (i < n4) ((float4*)p)[i] = make_float4(0.f, 0.f, 0.f, 0.f);
}

// ---- edge scatter-add: one wave32 per edge, float2 per lane ---------------
__global__ __launch_bounds__(256) void k_scatter(const float* __restrict__ xsrc,
                                                 const int* __restrict__ ei,
                                                 int ne,
                                                 float* __restrict__ agg) {
  int gid  = blockIdx.x * 256 + threadIdx.x;
  int e    = gid >> 5;
  int lane = gid & 31;
  if (e >= ne) return;
  int s = ei[e];           // ei layout: [2, ne] -> row 0 = src, row 1 = dst
  int d = ei[ne + e];
  float2 v = ((const float2*)(xsrc + (long)s * 64))[lane];
  float* dst = agg + (long)d * 64 + lane * 2;
  unsafeAtomicAdd(dst,     v.x);   // -> global_atomic_add_f32 (L2-resident)
  unsafeAtomicAdd(dst + 1, v.y);
}

// ---- WMMA fp32 GEMM:  Y[M x Nout] (+)= A@WA.T (+bias) (+ X@WX.T) ----------
// A, X: [M x 64] row-major.  WA, WX: [Nout x 64] row-major (so W.T is the
// K x N operand and B-fragments are contiguous float2 loads).  K = 64.
// One wave32 per 16-row tile; 8 waves (128 rows) per block.
__global__ __launch_bounds__(256)
void k_gemm(const float* __restrict__ A,  const float* __restrict__ WA,
            const float* __restrict__ X,  const float* __restrict__ WX,
            const float* __restrict__ bias, float* __restrict__ Y,
            int M, int Nout, int accum) {
  const int lane = threadIdx.x & 31;
  const int wave = threadIdx.x >> 5;
  const int m0   = (blockIdx.x * 8 + wave) * 16;
  if (m0 >= M) return;                      // wave-uniform: EXEC stays all-1s

  const int  lh    = lane & 15;             // row (A) / col (B,C)
  const int  kh    = lane >> 4;             // K sub-pair selector
  const long arow  = (long)(m0 + lh) * 64 + 2 * kh;   // A fragment base
  const int  crow0 = m0 + 8 * kh;                     // C/D row base

  for (int n0 = 0; n0 < Nout; n0 += 16) {
    const int   col  = n0 + lh;
    const long  brow = (long)col * 64 + 2 * kh;       // B fragment base
    const float bv   = bias ? bias[col] : 0.f;

    v8f c;
#pragma unroll
    for (int j = 0; j < 8; ++j)
      c[j] = (accum ? Y[(long)(crow0 + j) * Nout + col] : 0.f) + bv;

#pragma unroll
    for (int k0 = 0; k0 < 64; k0 += 4) {
      v2f a = *(const v2f*)(A  + arow + k0);
      v2f b = *(const v2f*)(WA + brow + k0);
      c = __builtin_amdgcn_wmma_f32_16x16x4_f32(false, a, false, b,
                                                (short)0, c, false, false);
    }
    if (X) {
#pragma unroll
      for (int k0 = 0; k0 < 64; k0 += 4) {
        v2f a = *(const v2f*)(X  + arow + k0);
        v2f b = *(const v2f*)(WX + brow + k0);
        c = __builtin_amdgcn_wmma_f32_16x16x4_f32(false, a, false, b,
                                                  (short)0, c, false, false);
      }
    }
#pragma unroll
    for (int j = 0; j < 8; ++j)
      Y[(long)(crow0 + j) * Nout + col] = c[j];
  }
}

// ---- leaky_relu (float4 granularity) --------------------------------------
__global__ __launch_bounds__(256) void k_leaky(const float* __restrict__ in,
                                               float* __restrict__ out, int n4) {
  int i = blockIdx.x * 256 + threadIdx.x;
  if (i >= n4) return;
  float4 v = ((const float4*)in)[i];
  v.x = v.x > 0.f ? v.x : 0.01f * v.x;
  v.y = v.y > 0.f ? v.y : 0.01f * v.y;
  v.z = v.z > 0.f ? v.z : 0.01f * v.z;
  v.w = v.w > 0.f ? v.w : 0.01f * v.w;
  ((float4*)out)[i] = v;
}

// ---------------------------------------------------------------------------
extern "C" void kernel_launch(void* const* d_in, const int* in_sizes, int n_in,
                              void* d_out, int out_size, void* d_ws, size_t ws_size,
                              hipStream_t stream) {
  const float* x_in[3] = { (const float*)d_in[0],    // ticker
                           (const float*)d_in[1],    // news
                           (const float*)d_in[2] };  // inst
  const int Ncnt[3] = { N_TICKER, N_NEWS, N_INST };

  // relations: {ei input idx, src node type, dst node type, sorted-param slot}
  struct Rel { int ein; int src; int dst; int pslot; };
  const Rel rels[5] = {
    { 3, 1, 0, 3 },  // mentions:     news   -> ticker
    { 4, 0, 1, 2 },  // mentioned_by: ticker -> news
    { 5, 2, 0, 1 },  // holds:        inst   -> ticker
    { 6, 0, 2, 0 },  // held_by:      ticker -> inst
    { 7, 0, 0, 4 },  // related:      ticker -> ticker
  };

  // workspace carve-up (floats)
  float* ws    = (float*)d_ws;
  float* agg   = ws;                      // 100000*64 = 6,400,000
  float* acc[3]; float* xb[3];
  acc[0] = agg    + 6400000;              // 3,200,000
  acc[1] = acc[0] + 3200000;              // 6,400,000
  acc[2] = acc[1] + 6400000;              //   640,000
  xb[0]  = acc[2] + 640000;
  xb[1]  = xb[0]  + 3200000;
  xb[2]  = xb[1]  + 6400000;

  const float* cur[3] = { x_in[0], x_in[1], x_in[2] };

  for (int l = 0; l < 2; ++l) {
    // zero per-dst accumulators
    for (int t = 0; t < 3; ++t) {
      int n4 = Ncnt[t] * 64 / 4;
      k_zero<<<(n4 + 255) / 256, 256, 0, stream>>>(acc[t], n4);
    }
    for (int r = 0; r < 5; ++r) {
      const Rel& R  = rels[r];
      const int  ne = in_sizes[R.ein] / 2;
      const int* ei = (const int*)d_in[R.ein];
      // zero agg for this dst
      int n4 = Ncnt[R.dst] * 64 / 4;
      k_zero<<<(n4 + 255) / 256, 256, 0, stream>>>(agg, n4);
      // scatter-add source rows into agg
      k_scatter<<<(ne + 7) / 8, 256, 0, stream>>>(cur[R.src], ei, ne, agg);
      // acc[dst] += agg@Wl.T + bl + x[dst]@Wr.T
      const int pb = 8 + l * 15 + R.pslot * 3;
      const float* Wl = (const float*)d_in[pb + 0];
      const float* Wr = (const float*)d_in[pb + 1];
      const float* bl = (const float*)d_in[pb + 2];
      const int tiles = Ncnt[R.dst] / 16;
      k_gemm<<<(tiles + 7) / 8, 256, 0, stream>>>(agg, Wl, cur[R.dst], Wr, bl,
                                                  acc[R.dst], Ncnt[R.dst], 64, 1);
    }
    // leaky_relu -> next-layer activations
    for (int t = 0; t < 3; ++t) {
      int n4 = Ncnt[t] * 64 / 4;
      k_leaky<<<(n4 + 255) / 256, 256, 0, stream>>>(acc[t], xb[t], n4);
      cur[t] = xb[t];
    }
  }

  // final: out = x_ticker @ lin_W.T + lin_b   (50000 x 32)
  const float* linW = (const float*)d_in[38];
  const float* linb = (const float*)d_in[39];
  const int tiles = N_TICKER / 16;
  k_gemm<<<(tiles + 7) / 8, 256, 0, stream>>>(cur[0], linW, nullptr, nullptr,
                                              linb, (float*)d_out,
                                              N_TICKER, 32, 0);
}